// LearnableShiftViews_25692494365146
// MI455X (gfx1250) — compile-verified
//
#include <hip/hip_runtime.h>
#include <stdint.h>

// LearnableShiftViews for MI455X (gfx1250).
// Memory-bound op (231 MB traffic, ~0.4 GFLOP) -> optimize the data path:
//   * stage each 224x224 fp32 plane (200,704 B) in LDS once via the Tensor
//     Data Mover (async tensor load, TENSORcnt), reuse for all 5 views
//   * fp32 VALU bilinear blend (weights are data-independent)
//   * non-temporal global_store_b128 for the write-once output stream

#define H_  224
#define W_  224
#define B_  64
#define C_  3
#define V_  5
#define HW_ (H_ * W_)

typedef float        v4f   __attribute__((ext_vector_type(4)));
typedef unsigned int u32x4 __attribute__((ext_vector_type(4)));
typedef int          i32x4 __attribute__((ext_vector_type(4)));
typedef int          i32x8 __attribute__((ext_vector_type(8)));

#ifndef __has_builtin
#define __has_builtin(x) 0
#endif

#if __has_builtin(__builtin_amdgcn_tensor_load_to_lds)
#define USE_TDM 1
#else
#define USE_TDM 0
#endif

#if USE_TDM
// Build a CDNA5 Tensor-DMA descriptor (CDNA5 ISA)
// for a 1-D copy of HW_ fp32 elements: global -> LDS.
__device__ __forceinline__ void tdm_load_plane(const float* gsrc, unsigned lds_byte_off) {
  const uint64_t ga = (uint64_t)(uintptr_t)gsrc;

  // ---- D# group 0 (128b): control + addresses ----
  u32x4 g0;
  g0[0] = 1u;                                   // count=1 (valid), user mode, no gather
  g0[1] = lds_byte_off;                         // lds_addr (bytes)
  g0[2] = (unsigned)(ga & 0xFFFFFFFFu);         // global_addr[31:0]
  g0[3] = (unsigned)((ga >> 32) & 0x01FFFFFFu)  // global_addr[56:32]
        | (2u << 30);                           // type=2 ("image")

  // ---- D# group 1 (256b): dims / tile ----
  // data_size=2 -> 4-byte elements; tensor_dim0 = tile_dim0 = HW_ (1-D tile).
  i32x8 g1;
  g1[0] = (int)(2u << 16);                        // data_size=4B; mask/pad/iterate=0
  g1[1] = (int)(((unsigned)HW_ & 0xFFFFu) << 16); // tensor_dim0[15:0]  (bits 63:48)
  g1[2] = (int)(1u << 16);                        // tensor_dim0[31:16]=0, tensor_dim1=1
  g1[3] = (int)(((unsigned)HW_ & 0xFFFFu) << 16); // tensor_dim1[31:16]=0, tile_dim0=HW_
  g1[4] = 0;                                      // tile_dim1=0 (1-D), tile_dim2=0
  g1[5] = (int)HW_;                               // tensor_dim0_stride[31:0]
  g1[6] = 0;                                      // stride hi / dim1_stride lo
  g1[7] = 0;                                      // dim1_stride hi

  const i32x4 z4 = {0, 0, 0, 0};                  // groups 2/3 unused (<=2D)

#if __has_include(<hip/amd_detail/amd_gfx1250_TDM.h>)
  const i32x8 z8 = {0, 0, 0, 0, 0, 0, 0, 0};
  __builtin_amdgcn_tensor_load_to_lds(g0, g1, z4, z4, z8, 0);   // clang-23 (6-arg)
#else
  __builtin_amdgcn_tensor_load_to_lds(g0, g1, z4, z4, 0);       // ROCm 7.2 (5-arg)
#endif

#if __has_builtin(__builtin_amdgcn_s_wait_tensorcnt)
  __builtin_amdgcn_s_wait_tensorcnt(0);
#else
  asm volatile("s_wait_tensorcnt 0x0" ::: "memory");
#endif
}
#endif // USE_TDM

__global__ __launch_bounds__(1024, 1) void shift_views_kernel(
    const float* __restrict__ x,      // [B, C, H, W]
    const float* __restrict__ offs,   // [V, 2]  (dy, dx)
    float* __restrict__ out)          // [V, B, C, H, W]
{
  __shared__ float plane[HW_];        // 200,704 B of the 320 KB/WGP LDS

  const int plane_id = blockIdx.x;    // b*C + c
  const int tid      = threadIdx.x;
  const int nthr     = blockDim.x;
  const float* src   = x + (size_t)plane_id * HW_;

#if USE_TDM
  if (tid < 32) {                     // wave 0 owns the DMA + TENSORcnt wait
    // Generic shared pointer's low 32 bits are the LDS byte address
    // (flat-aperture rule).
    const unsigned lds_off = (unsigned)(uintptr_t)&plane[0];
    tdm_load_plane(src, lds_off);
  }
#else
  for (int i = tid; i < HW_ / 4; i += nthr)
    ((v4f*)plane)[i] = ((const v4f*)src)[i];
#endif
  __syncthreads();

  constexpr float SY = (float)H_ / (float)(H_ - 1);
  constexpr float SX = (float)W_ / (float)(W_ - 1);
  constexpr int   QW = W_ / 4;        // quads per row  = 56
  constexpr int   NQ = HW_ / 4;       // quads per view = 12544

  for (int v = 0; v < V_; ++v) {
    const float dy = offs[2 * v + 0];
    const float dx = offs[2 * v + 1];
    float* __restrict__ op = out + ((size_t)v * (B_ * C_) + plane_id) * HW_;

    for (int q = tid; q < NQ; q += nthr) {
      const int h  = q / QW;
      const int w0 = (q - h * QW) * 4;

      // vertical blend weights (mask folded into weight, clamp for the gather)
      const float ypix = ((float)h + dy) * SY - 0.5f;
      const float fy   = floorf(ypix);
      float wy1 = ypix - fy;
      float wy0 = 1.0f - wy1;
      const int iy0 = (int)fy;
      const int iy1 = iy0 + 1;
      wy0 = (iy0 >= 0 && iy0 < H_) ? wy0 : 0.0f;
      wy1 = (iy1 >= 0 && iy1 < H_) ? wy1 : 0.0f;
      const int cy0 = min(max(iy0, 0), H_ - 1);
      const int cy1 = min(max(iy1, 0), H_ - 1);
      const float* __restrict__ r0 = &plane[cy0 * W_];
      const float* __restrict__ r1 = &plane[cy1 * W_];

      v4f res;
#pragma unroll
      for (int k = 0; k < 4; ++k) {
        const int   w    = w0 + k;
        const float xpix = ((float)w + dx) * SX - 0.5f;
        const float fx   = floorf(xpix);
        float wx1 = xpix - fx;
        float wx0 = 1.0f - wx1;
        const int ix0 = (int)fx;
        const int ix1 = ix0 + 1;
        wx0 = (ix0 >= 0 && ix0 < W_) ? wx0 : 0.0f;
        wx1 = (ix1 >= 0 && ix1 < W_) ? wx1 : 0.0f;
        const int cx0 = min(max(ix0, 0), W_ - 1);
        const int cx1 = min(max(ix1, 0), W_ - 1);

        const float t0 = r0[cx0] * wy0 + r1[cx0] * wy1;  // H blend
        const float t1 = r0[cx1] * wy0 + r1[cx1] * wy1;
        res[k] = t0 * wx0 + t1 * wx1;                    // W blend
      }
      // write-once output stream -> non-temporal B128 store
      __builtin_nontemporal_store(res, (v4f*)(op + (size_t)h * W_ + w0));
    }
  }
}

extern "C" void kernel_launch(void* const* d_in, const int* in_sizes, int n_in,
                              void* d_out, int out_size, void* d_ws, size_t ws_size,
                              hipStream_t stream) {
  (void)in_sizes; (void)n_in; (void)out_size; (void)d_ws; (void)ws_size;
  const float* x    = (const float*)d_in[0];
  const float* offs = (const float*)d_in[1];
  float*       out  = (float*)d_out;

  dim3 grid(B_ * C_);   // one workgroup per (b,c) plane -> 192 WGPs
  dim3 block(1024);     // 32 wave32 waves: fills a WGP
  hipLaunchKernelGGL(shift_views_kernel, grid, block, 0, stream, x, offs, out);
}